// GATConv__19335942766938
// MI455X (gfx1250) — compile-verified
//
#include <hip/hip_runtime.h>
#include <hip/hip_bf16.h>
#include <math.h>

typedef __attribute__((ext_vector_type(2))) float v2f;
typedef __attribute__((ext_vector_type(8))) float v8f;

#define IN_CH   128
#define HEADS   4
#define OUT_CH  32
#define NEG_SLOPE 0.2f
#define GAT_EPS 1e-16f

// Monotone float<->uint encoding so unsigned atomicMax == float max (handles negatives).
__device__ __forceinline__ unsigned fenc(float f) {
    unsigned u = __float_as_uint(f);
    return (u & 0x80000000u) ? ~u : (u | 0x80000000u);
}
__device__ __forceinline__ float fdec(unsigned e) {
    unsigned u = (e & 0x80000000u) ? (e & 0x7FFFFFFFu) : ~e;
    return __uint_as_float(u);
}
#define ENC_NEG_INF 0x007FFFFFu   // fenc(-inf)

// ---------------------------------------------------------------------------
// 0) init: zero out + denom, seed e_max with -inf
// ---------------------------------------------------------------------------
__global__ void gat_init(float* __restrict__ out, float* __restrict__ denom,
                         unsigned* __restrict__ emax, int N) {
    int i = blockIdx.x * blockDim.x + threadIdx.x;
    if (i < N * (HEADS * OUT_CH)) out[i] = 0.0f;
    if (i < N * HEADS) { denom[i] = 0.0f; emax[i] = ENC_NEG_INF; }
}

// ---------------------------------------------------------------------------
// 1) h = x @ W via V_WMMA_F32_16X16X4_F32 (fp32-exact).
//    One wave -> one 16x16 tile; 8 waves/block -> 16 rows x 128 cols.
//    A 16x4 layout: VGPR0 = K(+0 | +2 by lane-half), VGPR1 = K(+1 | +3).
//    B 4x16 layout: symmetric (row K striped across lanes, halves split K).
//    C/D: c[i] = D[M = i + 8*half][N = lane&15].
// ---------------------------------------------------------------------------
__global__ void gat_gemm_wmma(const float* __restrict__ x,
                              const float* __restrict__ W,
                              float* __restrict__ h) {
    const int lane  = threadIdx.x & 31;
    const int wave  = threadIdx.x >> 5;        // 0..7 -> N tile
    const int row0  = blockIdx.x * 16;
    const int col0  = wave * 16;
    const int half  = lane >> 4;
    const int lid   = lane & 15;

    const float* xrow = x + (size_t)(row0 + lid) * IN_CH;

    v8f acc = {};
    for (int k0 = 0; k0 < IN_CH; k0 += 4) {
        const int ka = k0 + 2 * half;
        v2f a, b;
        a.x = xrow[ka];
        a.y = xrow[ka + 1];
        b.x = W[(size_t)ka       * (HEADS * OUT_CH) + col0 + lid];
        b.y = W[(size_t)(ka + 1) * (HEADS * OUT_CH) + col0 + lid];
        acc = __builtin_amdgcn_wmma_f32_16x16x4_f32(
            /*neg_a=*/false, a, /*neg_b=*/false, b,
            /*c_mod=*/(short)0, acc, /*reuse_a=*/false, /*reuse_b=*/false);
    }
#pragma unroll
    for (int i = 0; i < 8; ++i)
        h[(size_t)(row0 + i + 8 * half) * (HEADS * OUT_CH) + col0 + lid] = acc[i];
}

// ---------------------------------------------------------------------------
// 2) per-node attention logits: a_src/a_dst [N,H]
// ---------------------------------------------------------------------------
__global__ void gat_logits(const float* __restrict__ h,
                           const float* __restrict__ att_src,
                           const float* __restrict__ att_dst,
                           float* __restrict__ a_src, float* __restrict__ a_dst,
                           int N) {
    int idx = blockIdx.x * blockDim.x + threadIdx.x;   // n*HEADS + hd
    if (idx >= N * HEADS) return;
    const int hd = idx & (HEADS - 1);
    const float* hp = h + (size_t)(idx >> 2) * (HEADS * OUT_CH) + hd * OUT_CH;
    const float* as = att_src + hd * OUT_CH;
    const float* ad = att_dst + hd * OUT_CH;
    float s = 0.0f, d = 0.0f;
#pragma unroll
    for (int c = 0; c < OUT_CH; ++c) { float v = hp[c]; s += v * as[c]; d += v * ad[c]; }
    a_src[idx] = s;
    a_dst[idx] = d;
}

__device__ __forceinline__ void edge_sd(const long long* ei, int i, int E,
                                        long long& s, long long& d) {
    if (i < E) { s = ei[i]; d = ei[(size_t)E + i]; }
    else       { s = d = (long long)(i - E); }          // self-loops
}

// ---------------------------------------------------------------------------
// 3) segment max of leaky-relu logits over dst
// ---------------------------------------------------------------------------
__global__ void gat_edge_max(const long long* __restrict__ ei,
                             const float* __restrict__ a_src,
                             const float* __restrict__ a_dst,
                             unsigned* __restrict__ emax, int E, int N) {
    int i = blockIdx.x * blockDim.x + threadIdx.x;
    if (i >= E + N) return;
    long long s, d; edge_sd(ei, i, E, s, d);
#pragma unroll
    for (int hd = 0; hd < HEADS; ++hd) {
        float e = a_src[(size_t)s * HEADS + hd] + a_dst[(size_t)d * HEADS + hd];
        e = (e >= 0.0f) ? e : NEG_SLOPE * e;
        atomicMax(&emax[(size_t)d * HEADS + hd], fenc(e));
    }
}

// ---------------------------------------------------------------------------
// 4) segment sum of exp(e - max) over dst
// ---------------------------------------------------------------------------
__global__ void gat_edge_den(const long long* __restrict__ ei,
                             const float* __restrict__ a_src,
                             const float* __restrict__ a_dst,
                             const unsigned* __restrict__ emax,
                             float* __restrict__ denom, int E, int N) {
    int i = blockIdx.x * blockDim.x + threadIdx.x;
    if (i >= E + N) return;
    long long s, d; edge_sd(ei, i, E, s, d);
#pragma unroll
    for (int hd = 0; hd < HEADS; ++hd) {
        float e = a_src[(size_t)s * HEADS + hd] + a_dst[(size_t)d * HEADS + hd];
        e = (e >= 0.0f) ? e : NEG_SLOPE * e;
        float ex = expf(e - fdec(emax[(size_t)d * HEADS + hd]));
        atomicAdd(&denom[(size_t)d * HEADS + hd], ex);
    }
}

// ---------------------------------------------------------------------------
// 5) weighted scatter-aggregate: wave per edge, lane = channel (coalesced)
// ---------------------------------------------------------------------------
__global__ void gat_aggregate(const long long* __restrict__ ei,
                              const float* __restrict__ h,
                              const float* __restrict__ a_src,
                              const float* __restrict__ a_dst,
                              const unsigned* __restrict__ emax,
                              const float* __restrict__ denom,
                              float* __restrict__ out, int E, int N) {
    int wid  = (blockIdx.x * blockDim.x + threadIdx.x) >> 5;
    int lane = threadIdx.x & 31;
    if (wid >= E + N) return;
    long long s, d; edge_sd(ei, wid, E, s, d);
#pragma unroll
    for (int hd = 0; hd < HEADS; ++hd) {
        float e = a_src[(size_t)s * HEADS + hd] + a_dst[(size_t)d * HEADS + hd];
        e = (e >= 0.0f) ? e : NEG_SLOPE * e;
        float alpha = expf(e - fdec(emax[(size_t)d * HEADS + hd])) /
                      (denom[(size_t)d * HEADS + hd] + GAT_EPS);
        float v = h[(size_t)s * (HEADS * OUT_CH) + hd * OUT_CH + lane];
        atomicAdd(&out[(size_t)d * (HEADS * OUT_CH) + hd * OUT_CH + lane], v * alpha);
    }
}

// ---------------------------------------------------------------------------
// 6) bias + relu (in place)
// ---------------------------------------------------------------------------
__global__ void gat_finalize(float* __restrict__ out, const float* __restrict__ bias, int n) {
    int i = blockIdx.x * blockDim.x + threadIdx.x;
    if (i >= n) return;
    float v = out[i] + bias[i & (HEADS * OUT_CH - 1)];
    out[i] = v > 0.0f ? v : 0.0f;
}

extern "C" void kernel_launch(void* const* d_in, const int* in_sizes, int n_in,
                              void* d_out, int out_size, void* d_ws, size_t ws_size,
                              hipStream_t stream) {
    const float*     x       = (const float*)d_in[0];
    const long long* ei      = (const long long*)d_in[1];     // int64 [2,E]
    const float*     W       = (const float*)d_in[2];
    const float*     att_src = (const float*)d_in[3];
    const float*     att_dst = (const float*)d_in[4];
    const float*     bias    = (const float*)d_in[5];
    float*           out     = (float*)d_out;

    const int N = in_sizes[0] / IN_CH;     // 50000
    const int E = in_sizes[1] / 2;         // 800000

    // workspace layout
    char* ws = (char*)d_ws;
    float*    h     = (float*)ws;                              ws += (size_t)N * HEADS * OUT_CH * 4;
    float*    a_src = (float*)ws;                              ws += (size_t)N * HEADS * 4;
    float*    a_dst = (float*)ws;                              ws += (size_t)N * HEADS * 4;
    unsigned* emax  = (unsigned*)ws;                           ws += (size_t)N * HEADS * 4;
    float*    denom = (float*)ws;

    const int B = 256;
    const int total_out = N * HEADS * OUT_CH;

    gat_init<<<(total_out + B - 1) / B, B, 0, stream>>>(out, denom, emax, N);

    gat_gemm_wmma<<<N / 16, B, 0, stream>>>(x, W, h);          // 8 waves/block = 16x128 stripe

    gat_logits<<<(N * HEADS + B - 1) / B, B, 0, stream>>>(h, att_src, att_dst, a_src, a_dst, N);

    const int EN = E + N;
    gat_edge_max<<<(EN + B - 1) / B, B, 0, stream>>>(ei, a_src, a_dst, emax, E, N);
    gat_edge_den<<<(EN + B - 1) / B, B, 0, stream>>>(ei, a_src, a_dst, emax, denom, E, N);

    gat_aggregate<<<(EN + 7) / 8, B, 0, stream>>>(ei, h, a_src, a_dst, emax, denom, out, E, N);

    gat_finalize<<<(total_out + B - 1) / B, B, 0, stream>>>(out, bias, total_out);
}